// Global_Refinement_15238543966316
// MI455X (gfx1250) — compile-verified
//
#include <hip/hip_runtime.h>

// ---------------------------------------------------------------------------
// Point-MLP (3 -> 128 -> 16) + permuted last-write-wins grid scatter.
//
// Strategy (MI455X, wave32, WMMA f32 16x16x4):
//  K1: init winner planes (channel-0 plane of d_out, bit-cast int) to -1
//  K2: atomicMax(winner[cell], j) over permuted positions j  (last write wins)
//  K3: per 16-cell tile: gather winning points, run MLP via v_wmma_f32_16x16x4_f32
//      (layer1: K=4 with bias folded; layer2: K=128 = 32 wmma steps via LDS
//       transpose), write all 16 channels (zeros for empty cells).
// No d_ws needed; everything deterministic per call.
// ---------------------------------------------------------------------------

#define B_    16
#define N_    131072
#define CMID  128
#define COUT  16
#define C32   32768                 // 32*32*32
#define C64   262144                // 64*64*64
#define OFF64 (B_ * COUT * C32)     // 8,388,608 floats: start of feat_cell64
#define TILES32      (B_ * C32 / 16)            // 32768
#define TILES_TOTAL  (B_ * (C32 + C64) / 16)    // 294912
#define WPB   4                     // waves per block (128 threads)

typedef __attribute__((ext_vector_type(2))) float v2f;
typedef __attribute__((ext_vector_type(8))) float v8f;

// ---- K1: winner planes = -1 ------------------------------------------------
__global__ void init_winner_kernel(int* outI) {
  int idx = blockIdx.x * blockDim.x + threadIdx.x;
  const int tot32 = B_ * C32;
  if (idx < tot32) {
    int b = idx / C32, c = idx - b * C32;
    outI[(size_t)b * COUT * C32 + c] = -1;
  } else {
    int k = idx - tot32;
    if (k < B_ * C64) {
      int b = k / C64, c = k - b * C64;
      outI[(size_t)OFF64 + (size_t)b * COUT * C64 + c] = -1;
    }
  }
}

// ---- K2: argmax-by-permuted-position scatter -------------------------------
__global__ void scatter_winner_kernel(const float* __restrict__ x,
                                      const int* __restrict__ perm,
                                      int* outI) {
  int tid = blockIdx.x * blockDim.x + threadIdx.x;
  if (tid >= B_ * N_) return;
  int b = tid / N_;
  int j = tid - b * N_;                 // position in permuted order
  int p = perm[tid];                    // original point id
  const float* xr = x + ((size_t)b * N_ + p) * 3;
  float xv = xr[0], yv = xr[1], zv = xr[2];

  int ix = (int)fminf(fmaxf(xv * 32.0f + 16.5f, 0.0f), 31.0f);
  int iy = (int)fminf(fmaxf(yv * 32.0f + 16.5f, 0.0f), 31.0f);
  int iz = (int)fminf(fmaxf(zv * 32.0f + 16.5f, 0.0f), 31.0f);
  int cell32 = (ix * 32 + iy) * 32 + iz;

  int jx = (int)fminf(fmaxf(xv * 64.0f + 32.5f, 0.0f), 63.0f);
  int jy = (int)fminf(fmaxf(yv * 64.0f + 32.5f, 0.0f), 63.0f);
  int jz = (int)fminf(fmaxf(zv * 64.0f + 32.5f, 0.0f), 63.0f);
  int cell64 = (jx * 64 + jy) * 64 + jz;

  atomicMax(outI + (size_t)b * COUT * C32 + cell32, j);
  atomicMax(outI + (size_t)OFF64 + (size_t)b * COUT * C64 + cell64, j);
}

// ---- K3: gather winners + MLP via WMMA -------------------------------------
__global__ __launch_bounds__(WPB * 32)
void mlp_gather_wmma_kernel(const float* __restrict__ x,
                            const float* __restrict__ W1,
                            const float* __restrict__ b1v,
                            const float* __restrict__ W2,
                            const float* __restrict__ b2v,
                            const int* __restrict__ perm,
                            float* out) {
  __shared__ __align__(16) float hs[WPB * 16 * CMID];  // per-wave h transpose, 32KB
  __shared__ __align__(16) float w1s[4 * CMID];        // W1 augmented (row3 = b1), 2KB
  __shared__ __align__(16) float w2s[COUT * CMID];     // W2 row-major, 8KB
  __shared__ int jbuf[WPB * 16];

  const int t = threadIdx.x;

  // Preload weights into LDS (B-operand-friendly layouts).
  for (int i = t; i < 4 * CMID; i += WPB * 32) {
    int k = i >> 7, c = i & (CMID - 1);
    w1s[i] = (k < 3) ? W1[c * 3 + k] : b1v[c];
  }
  for (int i = t; i < COUT * CMID; i += WPB * 32) w2s[i] = W2[i];

  const int lane = t & 31;
  const int wave = t >> 5;
  const int half = lane >> 4;   // 0: lanes 0-15, 1: lanes 16-31
  const int lm   = lane & 15;

  // Tile -> (grid, batch, cell range, output base).
  int T = blockIdx.x * WPB + wave;
  const int* winner;
  float* obase;
  int planeN, cell0, b;
  if (T < TILES32) {
    b      = T >> 11;                       // / (C32/16)
    cell0  = (T & 2047) << 4;
    winner = (const int*)out + (size_t)b * COUT * C32;
    obase  = out + (size_t)b * COUT * C32;
    planeN = C32;
  } else {
    int T2 = T - TILES32;
    b      = T2 >> 14;                      // / (C64/16)
    cell0  = (T2 & 16383) << 4;
    winner = (const int*)out + (size_t)OFF64 + (size_t)b * COUT * C64;
    obase  = out + (size_t)OFF64 + (size_t)b * COUT * C64;
    planeN = C64;
  }

  // Gather this lane's cell winner and its point coordinates.
  int j = winner[cell0 + lm];               // read channel-0 plane BEFORE overwrite
  if (lane < 16) jbuf[wave * 16 + lane] = j;
  int p = (j >= 0) ? perm[b * N_ + j] : 0;
  const float* xr = x + ((size_t)b * N_ + p) * 3;
  float xv = xr[0], yv = xr[1], zv = xr[2];

  // A operand, layer 1: 16 cells x K=4 (x,y,z,1). v0 -> K=2*half, v1 -> K=2*half+1.
  v2f a1;
  a1.x = half ? zv : xv;
  a1.y = half ? 1.0f : yv;

  __syncthreads();  // weights ready

  // ---- Layer 1: h[16x128] = relu(A1 * W1aug), 8 WMMAs ----
  v8f c1[8];
#pragma unroll
  for (int tt = 0; tt < 8; ++tt) {
    v2f bb;  // B: (K = 2*half + v, N = lm) of channel tile tt
    bb.x = w1s[(half * 2    ) * CMID + tt * 16 + lm];
    bb.y = w1s[(half * 2 + 1) * CMID + tt * 16 + lm];
    v8f z = {};
    c1[tt] = __builtin_amdgcn_wmma_f32_16x16x4_f32(
        false, a1, false, bb, (short)0, z, false, false);
  }

  // ReLU + transpose into LDS: hs[wave][point][channel].
  float* hw = hs + wave * (16 * CMID);
#pragma unroll
  for (int tt = 0; tt < 8; ++tt) {
#pragma unroll
    for (int r = 0; r < 8; ++r) {
      // D layout: (M = r + 8*half, N = lm) -> point, channel = 16*tt + lm
      hw[(r + 8 * half) * CMID + tt * 16 + lm] = fmaxf(c1[tt][r], 0.0f);
    }
  }
  __syncthreads();  // h (and jbuf) visible

  // ---- Layer 2: feat[16x16] = h * W2^T, K=128 in 32 WMMA steps ----
  v8f acc = {};
#pragma unroll
  for (int kk = 0; kk < 32; ++kk) {
    int ch = kk * 4 + half * 2;            // K-in-tile = 2*half + {0,1}
    v2f a2  = *(const v2f*)(hw  + lm * CMID + ch);  // A: (M=lm, K=ch..ch+1)
    v2f b2t = *(const v2f*)(w2s + lm * CMID + ch);  // B: (K=ch..ch+1, N=lm)
    acc = __builtin_amdgcn_wmma_f32_16x16x4_f32(
        false, a2, false, b2t, (short)0, acc, false, false);
  }
  float bias = b2v[lm];

  // Store: channel plane lm, cells cell0 + r + 8*half; zeros for empty cells.
#pragma unroll
  for (int r = 0; r < 8; ++r) {
    int m  = r + 8 * half;
    int jm = jbuf[wave * 16 + m];
    float val = (jm >= 0) ? (acc[r] + bias) : 0.0f;
    obase[(size_t)lm * planeN + cell0 + m] = val;
  }
}

// ---------------------------------------------------------------------------
extern "C" void kernel_launch(void* const* d_in, const int* in_sizes, int n_in,
                              void* d_out, int out_size, void* d_ws, size_t ws_size,
                              hipStream_t stream) {
  (void)in_sizes; (void)n_in; (void)out_size; (void)d_ws; (void)ws_size;
  const float* x    = (const float*)d_in[0];
  const float* W1   = (const float*)d_in[1];
  const float* b1   = (const float*)d_in[2];
  const float* W2   = (const float*)d_in[3];
  const float* b2   = (const float*)d_in[4];
  const int*   perm = (const int*)d_in[5];
  float* out = (float*)d_out;

  const int totWinner = B_ * (C32 + C64);          // 4,718,592
  init_winner_kernel<<<(totWinner + 255) / 256, 256, 0, stream>>>((int*)out);
  scatter_winner_kernel<<<(B_ * N_ + 255) / 256, 256, 0, stream>>>(x, perm, (int*)out);
  mlp_gather_wmma_kernel<<<TILES_TOTAL / WPB, WPB * 32, 0, stream>>>(
      x, W1, b1, W2, b2, perm, out);
}